// chamfer_lossv2_59261958750889
// MI455X (gfx1250) — compile-verified
//
#include <hip/hip_runtime.h>

// CDNA5 / gfx1250, wave32.
typedef __attribute__((ext_vector_type(2))) float v2f;
typedef __attribute__((ext_vector_type(8))) float v8f;

#define PTS   4096
#define NB    4
#define TILES 256                 // 4096/16 column tiles
#define PANEL_FLOATS 64           // 32 lanes * 2 floats per operand tile
#define STRIPS 2                  // row strips per wave
#define CHUNKS 4                  // n-loop split for parallelism
#define TILES_PER_CHUNK (TILES / CHUNKS)   // 64

// workspace layout (floats)
#define PACK_FLOATS (2 * NB * TILES * PANEL_FLOATS)   // 131072 per pack
#define APACK_OFF 0
#define BPACK_OFF PACK_FLOATS
#define MATCH_OFF (2 * PACK_FLOATS)
#define MATCH_STRIDE (2 * NB * PTS)                   // 32768 per chunk
// total: 262144 + 4*32768 = 393216 floats = 1.5 MB

// Direct v_min_num_f32 / v_min3_num_f32: bypasses llvm.minnum's mandatory
// v_max_num(x,x) sNaN-canonicalization of every WMMA result.
__device__ __forceinline__ float min2f(float a, float b) {
    float d;
    asm("v_min_num_f32 %0, %1, %2" : "=v"(d) : "v"(a), "v"(b));
    return d;
}
__device__ __forceinline__ float min3f(float a, float b, float c) {
    float d;
    asm("v_min3_num_f32 %0, %1, %2, %3" : "=v"(d) : "v"(a), "v"(b), "v"(c));
    return d;
}

// ---------------------------------------------------------------------------
// Kernel 1: pack A-panels (-2*p, 1) and B-panels (p, |p|^2) into the WMMA
// 16x4 / 4x16 f32 VGPR layout. pass 0: A=gt, B=coord. pass 1: swapped.
// Per ISA 7.12.2 (32-bit A 16x4): lanes 0-15 hold K=0,1 (V0,V1), lanes
// 16-31 hold K=2,3.  K = 2*hi + j for VGPR j.
// ---------------------------------------------------------------------------
__global__ void __launch_bounds__(256) cd_pack(const float* __restrict__ coord,
                                               const float* __restrict__ gt,
                                               float* __restrict__ ws) {
    int i    = blockIdx.x * blockDim.x + threadIdx.x;  // [0, 2*NB*TILES*32)
    int lane = i & 31;
    int s    = (i >> 5) & (TILES - 1);
    int b    = (i >> 13) & (NB - 1);
    int pass = i >> 15;
    int hi   = lane >> 4;
    int p    = s * 16 + (lane & 15);

    const float* srcA = (pass == 0) ? gt : coord;
    const float* srcB = (pass == 0) ? coord : gt;
    const float* pa = srcA + ((size_t)b * PTS + p) * 3;
    const float* pb = srcB + ((size_t)b * PTS + p) * 3;

    float a0 = pa[0], a1 = pa[1], a2 = pa[2];
    float b0 = pb[0], b1 = pb[1], b2 = pb[2];
    float bsq = b0 * b0 + b1 * b1 + b2 * b2;

    float av[4] = {-2.0f * a0, -2.0f * a1, -2.0f * a2, 1.0f};
    float bv[4] = {b0, b1, b2, bsq};

    size_t base = (size_t)i * 2;   // == (((pass*NB+b)*TILES+s)*32 + lane)*2
    float* ap = ws + APACK_OFF + base;
    float* bp = ws + BPACK_OFF + base;
    ap[0] = av[2 * hi + 0];
    ap[1] = av[2 * hi + 1];
    bp[0] = bv[2 * hi + 0];
    bp[1] = bv[2 * hi + 1];
}

// ---------------------------------------------------------------------------
// Kernel 2: each wave owns 2 row strips and one 64-tile column chunk.
// S = A x B via v_wmma_f32_16x16x4_f32 gives (dis - rownorm) for a 16x16
// tile; tiles processed in pairs, folded with v_min3_num_f32 into the
// running row-min accumulators.
// D-tile layout: lane L, VGPR r  ->  (m = r + 8*(L>=16), n = L%16).
// 4096 waves total.
// ---------------------------------------------------------------------------
__global__ void __launch_bounds__(256) cd_main(float* __restrict__ ws) {
    const int lane  = threadIdx.x & 31;
    const int w     = (blockIdx.x * blockDim.x + threadIdx.x) >> 5; // [0,4096)
    const int chunk = w & (CHUNKS - 1);
    const int sg    = (w >> 2) & (TILES / STRIPS - 1);  // strip group [0,128)
    const int b     = (w >> 9) & (NB - 1);
    const int pass  = w >> 11;

    const float* apanel = ws + APACK_OFF +
        (size_t)((pass * NB + b) * TILES + sg * STRIPS) * PANEL_FLOATS;
    const float* bpanel = ws + BPACK_OFF +
        (size_t)((pass * NB + b) * TILES + chunk * TILES_PER_CHUNK) * PANEL_FLOATS;

    v2f A[STRIPS];
#pragma unroll
    for (int i = 0; i < STRIPS; ++i) {
        const float* p = apanel + i * PANEL_FLOATS + lane * 2;
        A[i][0] = p[0];
        A[i][1] = p[1];
    }

    v8f mn[STRIPS];
#pragma unroll
    for (int i = 0; i < STRIPS; ++i)
#pragma unroll
        for (int r = 0; r < 8; ++r) mn[i][r] = 3.4e38f;

    const v8f cz = {};
    for (int t = 0; t < TILES_PER_CHUNK; t += 2) {
        const float* p0 = bpanel + (t + 0) * PANEL_FLOATS + lane * 2;
        const float* p1 = bpanel + (t + 1) * PANEL_FLOATS + lane * 2;
        v2f B0, B1;
        B0[0] = p0[0];
        B0[1] = p0[1];
        B1[0] = p1[0];
        B1[1] = p1[1];
#pragma unroll
        for (int i = 0; i < STRIPS; ++i) {
            v8f S0 = __builtin_amdgcn_wmma_f32_16x16x4_f32(
                false, A[i], false, B0, (short)0, cz, false, false);
            v8f S1 = __builtin_amdgcn_wmma_f32_16x16x4_f32(
                false, A[i], false, B1, (short)0, cz, false, false);
#pragma unroll
            for (int r = 0; r < 8; ++r) mn[i][r] = min3f(mn[i][r], S0[r], S1[r]);
        }
    }

    // Row-min: reduce across the 16 lanes of each half (xor masks < 16 stay
    // inside the half, so m = r + 8*hi is preserved).
#pragma unroll
    for (int i = 0; i < STRIPS; ++i) {
#pragma unroll
        for (int r = 0; r < 8; ++r) {
            float v = mn[i][r];
#pragma unroll
            for (int mask = 1; mask < 16; mask <<= 1)
                v = min2f(v, __shfl_xor(v, mask, 32));
            mn[i][r] = v;
        }
    }

    float* match = ws + MATCH_OFF +
        (size_t)chunk * MATCH_STRIDE + (size_t)(pass * NB + b) * PTS;
    const int hi = lane >> 4;
    if ((lane & 15) == 0) {   // lane 0 writes m0..+7, lane 16 writes m0+8..+15
#pragma unroll
        for (int i = 0; i < STRIPS; ++i) {
            int m0 = (sg * STRIPS + i) * 16 + hi * 8;
#pragma unroll
            for (int r = 0; r < 8; ++r) match[m0 + r] = mn[i][r];
        }
    }
}

// ---------------------------------------------------------------------------
// Kernel 3: combine chunk partial mins, then
// result = (sum(min over chunks) + sum(gt^2) + sum(coord^2)) / (B*PTS)
// (both chamfer means share the denominator B*PTS = 16384).
// ---------------------------------------------------------------------------
__global__ void __launch_bounds__(256) cd_reduce(const float* __restrict__ coord,
                                                 const float* __restrict__ gt,
                                                 const float* __restrict__ ws,
                                                 float* __restrict__ out) {
    const int tid = threadIdx.x;
    float acc = 0.0f;
    const float* mp = ws + MATCH_OFF;
    for (int i = tid; i < MATCH_STRIDE; i += 256) {
        float v = min3f(mp[i], mp[i + 1 * MATCH_STRIDE], mp[i + 2 * MATCH_STRIDE]);
        v = min2f(v, mp[i + 3 * MATCH_STRIDE]);
        acc += v;
    }
    for (int i = tid; i < NB * PTS * 3; i += 256) { float x = gt[i];    acc += x * x; }
    for (int i = tid; i < NB * PTS * 3; i += 256) { float x = coord[i]; acc += x * x; }

    __shared__ float red[256];
    red[tid] = acc;
    __syncthreads();
    for (int s = 128; s > 0; s >>= 1) {
        if (tid < s) red[tid] += red[tid + s];
        __syncthreads();
    }
    if (tid == 0) out[0] = 1.0f /*WEIGHT*/ * (red[0] / (float)(NB * PTS));
}

// ---------------------------------------------------------------------------
extern "C" void kernel_launch(void* const* d_in, const int* in_sizes, int n_in,
                              void* d_out, int out_size, void* d_ws, size_t ws_size,
                              hipStream_t stream) {
    const float* coord = (const float*)d_in[0];  // [B, N, 3] f32
    const float* gt    = (const float*)d_in[1];  // [B, M, 3] f32
    float* out = (float*)d_out;
    float* ws  = (float*)d_ws;

    // 1) pack operand panels: 2*NB*TILES*32 = 65536 threads
    cd_pack<<<256, 256, 0, stream>>>(coord, gt, ws);
    // 2) WMMA distance tiles + partial row-mins: 4096 waves = 512 blocks
    cd_main<<<512, 256, 0, stream>>>(ws);
    // 3) final scalar reduction
    cd_reduce<<<1, 256, 0, stream>>>(coord, gt, ws, out);
}